// SoFeatures_39427799777476
// MI455X (gfx1250) — compile-verified
//
#include <hip/hip_runtime.h>
#include <hip/hip_bf16.h>

typedef float v2f __attribute__((ext_vector_type(2)));
typedef float v8f __attribute__((ext_vector_type(8)));

#define MM      41
#define HALF    20
#define NPAIRS  1261          // #(m,n) with |m+n|<=20 ; sum(41-|m|)
#define NCOLS   164           // 41 k-values * 4 float components
#define ROW_TILES 79          // ceil(1261/16); last stripe overlaps (starts at 1245)
#define COL_TILES 11          // col starts 0,16,...,144,148 (last overlaps)
#define NBM     256           // B(128) * Nmodes(2)
#define TOTAL_STRIPES (NBM * ROW_TILES)   // 20224
#define WAVES_PER_BLOCK 8

// prefix(u) = # of (m,n) pairs with (m+20) < u   (u in 0..41)
__device__ __forceinline__ int pair_prefix(int u) {
    const int t  = u - 21;
    const int pa = (u * (u + 41)) >> 1;          // u <= 21
    const int pb = 651 + ((t * (81 - t)) >> 1);  // u >= 21 (agree at u==21)
    return (u <= 21) ? pa : pb;
}

// Per (b,mode): Out(1261 x 164) = A(1261 x 2) @ Bmat(2 x 164), K padded to 4 for
// V_WMMA_F32_16X16X4_F32.  One wave per 16-row stripe; overlapped edge tiles
// rewrite identical values so NO store needs a bounds check.
__global__ __launch_bounds__(256)
void so_features_wmma_kernel(const float* __restrict__ er,
                             const float* __restrict__ ei,
                             float* __restrict__ out)
{
    const int lane = threadIdx.x & 31;
    // readfirstlane -> compiler knows stripe & everything derived is wave-uniform
    const int wv     = __builtin_amdgcn_readfirstlane((int)(threadIdx.x >> 5));
    const int stripe = blockIdx.x * WAVES_PER_BLOCK + wv;

    const int rowTile = stripe % ROW_TILES;
    const int bm      = stripe / ROW_TILES;     // = b*2 + mode
    const int mode    = bm & 1;
    const int b       = bm >> 1;
    const int rowStart = min(rowTile * 16, NPAIRS - 16);   // overlapped last stripe

    const int half = lane >> 4;                 // lane half: K-pair {0,1} vs {2,3}
    const int sub  = lane & 15;
    const bool lo  = (half == 0);

    const float* erb = er + b * (MM * 2);       // E_real[b, i, md] at erb[i*2+md]
    const float* eib = ei + b * (MM * 2);

    // ---------- A fragment (16x4 f32): lanes 0-15 hold row M=sub, K={0,1}=E1;
    // lanes 16-31 hold K={2,3} = zero padding. ----------
    float a0, a1;
    {
        const int r = rowStart + sub;           // 0..1260, always valid
        // closed-form decode r -> u = m+20 via inverse prefix sum (+/-2 fixup)
        const float s1 = __builtin_sqrtf(1681.0f + 8.0f * (float)r);
        const int   u1 = (int)((s1 - 41.0f) * 0.5f);
        const int   J  = r - 651;
        const float s2 = __builtin_sqrtf(6561.0f - 8.0f * (float)J);
        const int   u2 = 21 + (int)((81.0f - s2) * 0.5f);
        int u = (r < 651) ? u1 : u2;
        u += (pair_prefix(u + 1) <= r);
        u += (pair_prefix(u + 1) <= r);
        u -= (pair_prefix(u) > r);
        u -= (pair_prefix(u) > r);
        const int rem = r - pair_prefix(u);
        const int m   = u - HALF;
        const int n   = ((m >= 0) ? -HALF : (-HALF - m)) + rem;
        const int i_m = HALF + m, i_n = HALF + n, i_mn = HALF + m + n;

        const float emr  = erb[i_m * 2 + mode],  emi  = eib[i_m * 2 + mode];
        const float enr  = erb[i_n * 2 + mode],  eni  = eib[i_n * 2 + mode];
        const float emnr = erb[i_mn * 2 + mode], emni = eib[i_mn * 2 + mode];
        const float tr  = emr * enr - emi * eni;      // Em*En
        const float ti  = emr * eni + emi * enr;
        const float e1r = tr * emnr + ti * emni;      // * conj(Emn)
        const float e1i = ti * emnr - tr * emni;
        a0 = lo ? e1r : 0.f;
        a1 = lo ? e1i : 0.f;
    }
    const v2f A = {a0, a1};

    // Per-lane stripe base: element (v, colStart) at p[v*164 + colStart]
    float* const p = out + (size_t)bm * ((size_t)NPAIRS * NCOLS)
                         + (size_t)(rowStart + 8 * half) * NCOLS + sub;

    #pragma unroll
    for (int ct = 0; ct < COL_TILES; ++ct) {
        const int colStart = (ct < COL_TILES - 1) ? ct * 16 : (NCOLS - 16); // 148
        // ---------- B fragment (4x16 f32): lanes 0-15 carry K rows 0,1 for
        // col N=sub of this tile; lanes 16-31 = zero padding. ----------
        const int c  = colStart + sub;          // always < 164
        const int kk = c >> 2;                  // k index 0..40
        const int cc = c & 3;
        float t1 = 0.f, t2r = 0.f, t2i = 0.f;
        #pragma unroll
        for (int md = 0; md < 2; ++md) {
            const float ekr  = erb[kk * 2 + md],        eki  = eib[kk * 2 + md];
            const float enkr = erb[(40 - kk) * 2 + md], enki = eib[(40 - kk) * 2 + md];
            t1  += ekr * ekr  + eki * eki;      // sum_modes |Ek|^2
            t2r += ekr * enkr - eki * enki;     // sum_modes Ek*Enk
            t2i += ekr * enki + eki * enkr;
        }
        const float v0 = (cc == 0) ? t1 : (cc == 1) ? 0.f : (cc == 2) ? t2r : t2i;
        const float v1 = (cc == 0) ? 0.f : (cc == 1) ? t1 : (cc == 2) ? t2i : -t2r;
        const v2f B = {lo ? v0 : 0.f, lo ? v1 : 0.f};

        v8f C = {};
        // (neg_a, A, neg_b, B, c_mod, C, reuse_a, reuse_b)
        const v8f D = __builtin_amdgcn_wmma_f32_16x16x4_f32(false, A, false, B,
                                                            (short)0, C, false, false);

        // 8 unconditional NT stores, immediate offsets only (max 5184 B).
        #pragma unroll
        for (int v = 0; v < 8; ++v)
            __builtin_nontemporal_store(D[v], p + v * NCOLS + colStart);
    }
}

extern "C" void kernel_launch(void* const* d_in, const int* in_sizes, int n_in,
                              void* d_out, int out_size, void* d_ws, size_t ws_size,
                              hipStream_t stream) {
    (void)in_sizes; (void)n_in; (void)out_size; (void)d_ws; (void)ws_size;
    const float* er = (const float*)d_in[0];   // E_real (128,41,2) f32
    const float* ei = (const float*)d_in[1];   // E_imag (128,41,2) f32
    float* out = (float*)d_out;                // complex64 as interleaved f32
    const int blocks = TOTAL_STRIPES / WAVES_PER_BLOCK;   // 2528, exact
    so_features_wmma_kernel<<<blocks, 256, 0, stream>>>(er, ei, out);
}